// SAGE_33182917328949
// MI455X (gfx1250) — compile-verified
//
#include <hip/hip_runtime.h>
#include <hip/hip_bf16.h>

// ---------------------------------------------------------------------------
// GraphSAGE 2-layer forward for MI455X (gfx1250, wave32).
//   layer:  D = (agg/deg) @ W_l^T + b + x @ W_r^T      (fused into one WMMA chain)
//   matrix op: V_WMMA_F32_16X16X4_F32 (full fp32 — GEMM is ~4.5 GFLOP total,
//   far below any WMMA roofline; the kernel is aggregation/L2-atomic bound,
//   so no precision downcast is warranted).
// ---------------------------------------------------------------------------

typedef __attribute__((ext_vector_type(2))) float v2f;
typedef __attribute__((ext_vector_type(8))) float v8f;

#define KDIM 128   // IN == HID == 128 for both layers
#define BN_EPS 1e-5f

__device__ __forceinline__ void atomic_add_f32(float* p, float v) {
    __hip_atomic_fetch_add(p, v, __ATOMIC_RELAXED, __HIP_MEMORY_SCOPE_AGENT);
}

// ---------------------------------------------------------------------------
// Mean-aggregation scatter: one wave (32 lanes) per edge; each lane moves a
// float4 chunk of the 128-float source row into the destination row with
// hardware fp32 global atomics. Lane/chunk 0 also bumps the degree counter.
// x (25.6 MB) and agg (25.6 MB) are L2-resident on MI455X (192 MB L2).
// ---------------------------------------------------------------------------
__global__ void sage_scatter_kernel(const float* __restrict__ x,
                                    const int* __restrict__ src,
                                    const int* __restrict__ dst,
                                    float* __restrict__ agg,
                                    float* __restrict__ deg,
                                    int nedges) {
    const int chunks = KDIM / 4;                       // 32 = one wave
    long long tid = (long long)blockIdx.x * blockDim.x + threadIdx.x;
    long long e = tid / chunks;
    if (e >= nedges) return;
    int ch = (int)(tid % chunks);
    int s = src[e];
    int d = dst[e];
    const float4 v = *(const float4*)(x + (long long)s * KDIM + ch * 4);
    float* ap = agg + (long long)d * KDIM + ch * 4;
    atomic_add_f32(ap + 0, v.x);
    atomic_add_f32(ap + 1, v.y);
    atomic_add_f32(ap + 2, v.z);
    atomic_add_f32(ap + 3, v.w);
    if (ch == 0) atomic_add_f32(deg + d, 1.0f);
}

__global__ void rdeg_kernel(const float* __restrict__ deg,
                            float* __restrict__ rdeg, int n) {
    int i = blockIdx.x * blockDim.x + threadIdx.x;
    if (i < n) rdeg[i] = 1.0f / fmaxf(deg[i], 1.0f);
}

// ---------------------------------------------------------------------------
// Fused SAGE GEMM:  D[M,Nout] = (A1 * rdeg_row) @ Wl^T + bias + A2 @ Wr^T
// One wave per 16x16 output tile; K=128 fully unrolled in steps of 4 with
// V_WMMA_F32_16X16X4_F32, two back-to-back WMMA chains into one accumulator.
//
// GUARD=false (layer 1, Nout=128): no column guard at all -> branch-free loop.
// GUARD=true  (layer 2, Nout=47):  clamped address + VALU mask multiply
// instead of divergent loads -> no EXEC save/restore in the inner loop.
//
// Fragment layouts (ISA 7.12.2, wave32):
//   A 16x4 f32 : v0/v1; lanes 0-15 -> (M=lane, K=k0/k0+1),
//                        lanes 16-31 -> (M=lane-16, K=k0+2/k0+3)
//   B 4x16 f32 : mirrored (K in the same slots, N = lane%16)
//   C/D 16x16  : VGPR r, lanes 0-15 -> (M=r, N=lane),
//                        lanes 16-31 -> (M=r+8, N=lane-16)
// ---------------------------------------------------------------------------
template <bool GUARD>
__global__ void sage_gemm_kernel(const float* __restrict__ A1,   // aggregated [M,K]
                                 const float* __restrict__ rdeg, // [M]
                                 const float* __restrict__ A2,   // root feats [M,K]
                                 const float* __restrict__ Wl,   // [Nout,K]
                                 const float* __restrict__ Wr,   // [Nout,K]
                                 const float* __restrict__ bias, // [Nout]
                                 float* __restrict__ D,          // [M,Nout]
                                 int Nout, int n_tiles_total, int n_tiles_col) {
    int wave = (blockIdx.x * blockDim.x + threadIdx.x) >> 5;   // wave32
    if (wave >= n_tiles_total) return;                          // wave-uniform
    int lane = threadIdx.x & 31;
    int l    = lane & 15;
    int hi   = lane >> 4;

    int tm = wave / n_tiles_col;
    int tn = wave % n_tiles_col;
    int M0 = tm * 16;
    int N0 = tn * 16;

    int row = M0 + l;
    int col = N0 + l;

    // Column handling without divergence: clamp the address, mask the value.
    int   colc  = GUARD ? min(col, Nout - 1) : col;
    float cmask = (!GUARD || col < Nout) ? 1.0f : 0.0f;

    float bv = bias[colc] * cmask;
    v8f c;
#pragma unroll
    for (int r = 0; r < 8; ++r) c[r] = bv;

    float rs = rdeg[row];
    const float* a1p = A1 + (long long)row * KDIM;
    const float* a2p = A2 + (long long)row * KDIM;
    const float* b1p = Wl + (long long)colc * KDIM;
    const float* b2p = Wr + (long long)colc * KDIM;

    const int kh = 2 * hi;
#pragma unroll
    for (int k0 = 0; k0 < KDIM; k0 += 4) {
        int ka = k0 + kh;

        v2f a1 = *(const v2f*)(a1p + ka);
        a1.x *= rs;
        a1.y *= rs;
        v2f b1 = *(const v2f*)(b1p + ka);
        if (GUARD) { b1.x *= cmask; b1.y *= cmask; }
        c = __builtin_amdgcn_wmma_f32_16x16x4_f32(false, a1, false, b1,
                                                  (short)0, c, false, false);

        v2f a2 = *(const v2f*)(a2p + ka);
        v2f b2 = *(const v2f*)(b2p + ka);
        if (GUARD) { b2.x *= cmask; b2.y *= cmask; }
        c = __builtin_amdgcn_wmma_f32_16x16x4_f32(false, a2, false, b2,
                                                  (short)0, c, false, false);
    }

    if (!GUARD || col < Nout) {
#pragma unroll
        for (int r = 0; r < 8; ++r) {
            int orow = M0 + r + 8 * hi;
            D[(long long)orow * Nout + col] = c[r];
        }
    }
}

// ---------------------------------------------------------------------------
// BatchNorm statistics: per-feature sum and sum-of-squares over N rows.
// blockDim.x == 128 (one thread per feature), grid-strided over rows,
// one pair of atomics per thread per block.
// ---------------------------------------------------------------------------
__global__ void bn_stats_kernel(const float* __restrict__ h,
                                float* __restrict__ sums,   // [128]
                                float* __restrict__ sumsq,  // [128]
                                int n) {
    int f = threadIdx.x;
    float s = 0.0f, s2 = 0.0f;
    for (int row = blockIdx.x; row < n; row += gridDim.x) {
        float v = h[(long long)row * KDIM + f];
        s += v;
        s2 += v * v;
    }
    atomic_add_f32(&sums[f], s);
    atomic_add_f32(&sumsq[f], s2);
}

__global__ void bn_apply_relu_kernel(const float* __restrict__ h_pre,
                                     const float* __restrict__ sums,
                                     const float* __restrict__ sumsq,
                                     const float* __restrict__ gamma,
                                     const float* __restrict__ beta,
                                     float* __restrict__ h_out,
                                     int n, float inv_n) {
    long long tid = (long long)blockIdx.x * blockDim.x + threadIdx.x;
    if (tid >= (long long)n * KDIM) return;
    int f = (int)(tid % KDIM);
    float mu = sums[f] * inv_n;
    float var = sumsq[f] * inv_n - mu * mu;
    float scale = __frsqrt_rn(var + BN_EPS) * gamma[f];
    float v = (h_pre[tid] - mu) * scale + beta[f];
    h_out[tid] = fmaxf(v, 0.0f);
}

// ---------------------------------------------------------------------------
extern "C" void kernel_launch(void* const* d_in, const int* in_sizes, int n_in,
                              void* d_out, int out_size, void* d_ws, size_t ws_size,
                              hipStream_t stream) {
    const float* x     = (const float*)d_in[0];
    const int*   edges = (const int*)d_in[1];
    const float* W1_l  = (const float*)d_in[2];
    const float* b1    = (const float*)d_in[3];
    const float* W1_r  = (const float*)d_in[4];
    const float* gamma = (const float*)d_in[5];
    const float* beta  = (const float*)d_in[6];
    const float* W2_l  = (const float*)d_in[7];
    const float* b2    = (const float*)d_in[8];
    const float* W2_r  = (const float*)d_in[9];

    const int N   = in_sizes[0] / KDIM;       // 50000
    const int E   = in_sizes[1] / 2;          // 800000
    const int OUT = in_sizes[7] / KDIM;       // 47

    const int* src = edges;
    const int* dst = edges + E;

    // d_out = [logit (N*OUT) | h (N*KDIM)]
    float* logit = (float*)d_out;
    float* h_out = logit + (long long)N * OUT;

    // workspace layout
    float* aggA  = (float*)d_ws;                       // N*KDIM
    float* h_pre = aggA + (size_t)N * KDIM;            // N*KDIM
    float* deg   = h_pre + (size_t)N * KDIM;           // N
    float* rdeg  = deg + N;                            // N
    float* sums  = rdeg + N;                           // KDIM
    float* sumsq = sums + KDIM;                        // KDIM

    const long long scatter_threads = (long long)E * (KDIM / 4);
    const int scatter_blocks = (int)((scatter_threads + 255) / 256);
    const int rdeg_blocks = (N + 255) / 256;

    // ---------------- Layer 1 ----------------
    hipMemsetAsync(aggA, 0, (size_t)N * KDIM * sizeof(float), stream);
    hipMemsetAsync(deg, 0, (size_t)N * sizeof(float), stream);
    hipMemsetAsync(sums, 0, 2 * KDIM * sizeof(float), stream);

    sage_scatter_kernel<<<scatter_blocks, 256, 0, stream>>>(x, src, dst, aggA, deg, E);
    rdeg_kernel<<<rdeg_blocks, 256, 0, stream>>>(deg, rdeg, N);

    {
        int n_tiles_col = KDIM / 16;                       // 8
        int n_tiles = (N / 16) * n_tiles_col;              // 25000
        int blocks = (n_tiles + 3) / 4;                    // 4 waves / 128-thread block
        sage_gemm_kernel<false><<<blocks, 128, 0, stream>>>(aggA, rdeg, x, W1_l, W1_r, b1,
                                                            h_pre, KDIM, n_tiles, n_tiles_col);
    }

    bn_stats_kernel<<<512, KDIM, 0, stream>>>(h_pre, sums, sumsq, N);
    {
        long long tot = (long long)N * KDIM;
        int blocks = (int)((tot + 255) / 256);
        bn_apply_relu_kernel<<<blocks, 256, 0, stream>>>(h_pre, sums, sumsq, gamma, beta,
                                                         h_out, N, 1.0f / (float)N);
    }

    // ---------------- Layer 2 ----------------
    hipMemsetAsync(aggA, 0, (size_t)N * KDIM * sizeof(float), stream);
    hipMemsetAsync(deg, 0, (size_t)N * sizeof(float), stream);

    sage_scatter_kernel<<<scatter_blocks, 256, 0, stream>>>(h_out, src, dst, aggA, deg, E);
    rdeg_kernel<<<rdeg_blocks, 256, 0, stream>>>(deg, rdeg, N);

    {
        int n_tiles_col = (OUT + 15) / 16;                 // 3
        int n_tiles = (N / 16) * n_tiles_col;              // 9375
        int blocks = (n_tiles + 3) / 4;
        sage_gemm_kernel<true><<<blocks, 128, 0, stream>>>(aggA, rdeg, h_out, W2_l, W2_r, b2,
                                                           logit, OUT, n_tiles, n_tiles_col);
    }
}